// PeriodicityTransform_4492535791822
// MI455X (gfx1250) — compile-verified
//
#include <hip/hip_runtime.h>
#include <math.h>

typedef float v2f __attribute__((ext_vector_type(2)));
typedef float v8f __attribute__((ext_vector_type(8)));

#define T_LEN   512
#define BN_SEQ  2048
#define NFREQ   256   // bins 1..256 (DC excluded; reference zeroes it)
#define KP      4
#define P_MAX   64
#define C_MAX   64
#define P_MIN   8

// ---------------------------------------------------------------------------
// Kernel 1: transpose x[B=32, T=512, N=64] -> seqs[BN=2048, T=512] (row-major)
// ---------------------------------------------------------------------------
__global__ void k_prep_seqs(const float* __restrict__ x, float* __restrict__ seqs) {
    int idx = blockIdx.x * blockDim.x + threadIdx.x;     // 0 .. BN*T-1
    int bn  = idx >> 9;                                  // /512
    int t   = idx & 511;
    int b   = bn >> 6;                                   // /64
    int n   = bn & 63;
    seqs[idx] = x[(b * T_LEN + t) * 64 + n];
}

// ---------------------------------------------------------------------------
// Kernel 2: DFT basis, row-major [t=0..511][col=0..511]
//   col < 256 : cos(-2*pi*(col+1)*t/512)   (real part)
//   col >= 256: sin(-2*pi*(col-255)*t/512) (imag part; sign irrelevant, squared)
// Exact integer phase reduction: (f*t) mod 512.
// ---------------------------------------------------------------------------
__global__ void k_prep_basis(float* __restrict__ basis) {
    int idx = blockIdx.x * blockDim.x + threadIdx.x;     // 0 .. 512*512-1
    int t   = idx >> 9;
    int c   = idx & 511;
    int f   = (c & 255) + 1;
    int ph  = (f * t) & 511;                             // exact mod 512
    float ang = -6.28318530717958647692f * (float)ph * (1.0f / 512.0f);
    basis[idx] = (c < 256) ? cosf(ang) : sinf(ang);
}

// ---------------------------------------------------------------------------
// Kernel 3: WMMA DFT.  mag2[bn][f] = (seqs@cos)^2 + (seqs@sin)^2
// One wave per 16(M) x 16(freq) tile; K=512 in steps of 4 via
// V_WMMA_F32_16X16X4_F32, two accumulators (cos tile, sin tile).
// Fragment layouts per CDNA5 ISA 7.12.2:
//   A 16x4 f32 : lane L holds A[L%16, 2*(L/16)+v], v in {0,1}
//   B 4x16 f32 : lane L holds B[2*(L/16)+v, L%16]
//   C/D 16x16  : lane L, vgpr r -> (M = r + 8*(L/16), N = L%16)
// ---------------------------------------------------------------------------
__global__ void k_dft_wmma(const float* __restrict__ seqs,
                           const float* __restrict__ basis,
                           float* __restrict__ mag2) {
    int wave  = (blockIdx.x * blockDim.x + threadIdx.x) >> 5;  // global wave id
    int lane  = threadIdx.x & 31;
    int mtile = wave >> 4;            // 0..127  (2048 rows / 16)
    int ftile = wave & 15;            // 0..15   (256 freqs / 16)
    int half  = lane >> 4;            // 0 or 1
    int lrow  = lane & 15;

    int arow = mtile * 16 + lrow;     // A-matrix row (M) for this lane
    int bcol = ftile * 16 + lrow;     // freq column (N) for this lane

    const float* __restrict__ ap = seqs + arow * T_LEN;

    v8f acc_re = {};
    v8f acc_im = {};

    for (int kk = 0; kk < T_LEN; kk += 4) {
        int k0 = kk + (half << 1);
        v2f a, bc, bs;
        a.x = ap[k0];
        a.y = ap[k0 + 1];
        const float* __restrict__ br0 = basis + k0 * 512;
        const float* __restrict__ br1 = basis + (k0 + 1) * 512;
        bc.x = br0[bcol];        bc.y = br1[bcol];          // cos columns
        bs.x = br0[256 + bcol];  bs.y = br1[256 + bcol];    // sin columns
        // (neg_a, A, neg_b, B, c_mod, C, reuse_a, reuse_b)
        acc_re = __builtin_amdgcn_wmma_f32_16x16x4_f32(
            false, a, false, bc, (short)0, acc_re, false, false);
        acc_im = __builtin_amdgcn_wmma_f32_16x16x4_f32(
            false, a, false, bs, (short)0, acc_im, false, false);
    }

    int fcol = ftile * 16 + lrow;
#pragma unroll
    for (int r = 0; r < 8; ++r) {
        int m = mtile * 16 + r + 8 * half;
        float re = acc_re[r];
        float im = acc_im[r];
        mag2[m * NFREQ + fcol] = re * re + im * im;
    }
}

// ---------------------------------------------------------------------------
// Kernel 4: top-4 per sequence (strict '>' scan, lowest index wins on ties,
// matching jax.lax.top_k), then period/cycle computation.
// Ints kept in ws for the gather; float copies written into d_out tail.
// ---------------------------------------------------------------------------
__global__ void k_topk(const float* __restrict__ mag2,
                       int* __restrict__ periods_i,
                       int* __restrict__ cycles_i,
                       float* __restrict__ out_pc) {  // out_pc: [periods(8192) | cycles(8192)]
    int bn = blockIdx.x * blockDim.x + threadIdx.x;
    if (bn >= BN_SEQ) return;
    const float* m = mag2 + bn * NFREQ;
    int best[KP];
#pragma unroll
    for (int s = 0; s < KP; ++s) {
        float bv = -1.0f;
        int   bi = 0;
        for (int f = 0; f < NFREQ; ++f) {
            bool taken = false;
            for (int q = 0; q < s; ++q) taken = taken || (best[q] == f);
            float v = m[f];
            if (!taken && v > bv) { bv = v; bi = f; }
        }
        best[s] = bi;
        int freq   = bi + 1;                 // bin index in full rfft spectrum
        int period = T_LEN / freq;
        if (period < P_MIN) period = P_MIN;
        if (period > P_MAX) period = P_MAX;
        int cyc = T_LEN / period;
        periods_i[bn * KP + s] = period;
        cycles_i [bn * KP + s] = cyc;
        out_pc[bn * KP + s]                 = (float)period;
        out_pc[BN_SEQ * KP + bn * KP + s]   = (float)cyc;
    }
}

// ---------------------------------------------------------------------------
// Kernel 5: gather + zero-pad. One block per (bn,k) 64x64 tile; writes every
// element (zeros outside the valid C x P window) -> d_out fully initialized.
// Bandwidth-bound: 128 MB of stores, coalesced.
// ---------------------------------------------------------------------------
__global__ void k_gather(const float* __restrict__ seqs,
                         const int* __restrict__ periods_i,
                         const int* __restrict__ cycles_i,
                         float* __restrict__ out) {
    int bnk = blockIdx.x;                    // 0 .. BN*KP-1
    int bn  = bnk >> 2;
    int P   = periods_i[bnk];
    int C   = cycles_i[bnk];
    int start = T_LEN - C * P;
    const float* __restrict__ s = seqs + bn * T_LEN;
    float* __restrict__ o = out + (size_t)bnk * (C_MAX * P_MAX);
    for (int i = threadIdx.x; i < C_MAX * P_MAX; i += blockDim.x) {
        int c = i >> 6;
        int p = i & 63;
        float v = 0.0f;
        if (c < C && p < P) v = s[start + c * P + p];
        o[i] = v;
    }
}

// ---------------------------------------------------------------------------
extern "C" void kernel_launch(void* const* d_in, const int* in_sizes, int n_in,
                              void* d_out, int out_size, void* d_ws, size_t ws_size,
                              hipStream_t stream) {
    const float* x = (const float*)d_in[0];
    float* out = (float*)d_out;

    // ws layout (floats): seqs | basis | mag2 | periods(int) | cycles(int)
    float* ws    = (float*)d_ws;
    float* seqs  = ws;                                  // 2048*512
    float* basis = seqs  + BN_SEQ * T_LEN;              // 512*512
    float* mag2  = basis + T_LEN * 512;                 // 2048*256
    int*   periods_i = (int*)(mag2 + BN_SEQ * NFREQ);   // 2048*4
    int*   cycles_i  = periods_i + BN_SEQ * KP;         // 2048*4

    const size_t values_elems = (size_t)BN_SEQ * KP * C_MAX * P_MAX;  // 33,554,432

    k_prep_seqs <<<(BN_SEQ * T_LEN) / 256, 256, 0, stream>>>(x, seqs);
    k_prep_basis<<<(T_LEN * 512) / 256,    256, 0, stream>>>(basis);
    // 2048 tiles, 8 waves (256 threads) per block -> 256 blocks
    k_dft_wmma  <<<(BN_SEQ / 16) * (NFREQ / 16) / 8, 256, 0, stream>>>(seqs, basis, mag2);
    k_topk      <<<BN_SEQ / 256, 256, 0, stream>>>(mag2, periods_i, cycles_i,
                                                   out + values_elems);
    k_gather    <<<BN_SEQ * KP, 256, 0, stream>>>(seqs, periods_i, cycles_i, out);
}